// apexTransducerLoss_33011118637266
// MI455X (gfx1250) — compile-verified
//
#include <hip/hip_runtime.h>
#include <hip/hip_bf16.h>

// RNN-T transducer loss forward for B=16, T=192, U1=64, V=1024 (fp32).
// Phase 1 (memory-bound, 805MB stream): per-row logsumexp over V + gather of
//   blank/label logits. One wave32 per row; WMMA(f32 16x16x4, B=ones) performs
//   the cross-lane exp-sum reduction on the matrix unit.
// Phase 2: anti-diagonal wavefront over the 192x64 alpha lattice, 64 threads/batch.
// Phase 3: -mean(ll) scalar.

#define RB 16
#define RT 192
#define RU1 64
#define RU 63
#define RV 1024

typedef __attribute__((ext_vector_type(4))) float f32x4;
typedef __attribute__((ext_vector_type(2))) float v2f;
typedef __attribute__((ext_vector_type(8))) float v8f;

// ---------------------------------------------------------------------------
// Phase 1: one wave per (b,t,u) row of V=1024 logits.
//   lse = m + log(sum exp(x-m));  blank = x[0]-lse;  label = x[y[b,u]]-lse.
// ---------------------------------------------------------------------------
__global__ __launch_bounds__(256)
void rnnt_lse_kernel(const float* __restrict__ logits,
                     const int* __restrict__ y,
                     float* __restrict__ blank_lp,
                     float* __restrict__ label_lp) {
  const int lane = threadIdx.x & 31;
  const int row  = blockIdx.x * (blockDim.x >> 5) + (threadIdx.x >> 5);

  const float* rp  = logits + (size_t)row * RV;
  const f32x4* rp4 = (const f32x4*)rp;

  // 8 x b128 per lane: lane l, chunk c covers float4 index c*32+l (coalesced,
  // 512B per instruction across the wave). Non-temporal: 805MB single-use stream.
  f32x4 x[8];
#pragma unroll
  for (int c = 0; c < 8; ++c)
    x[c] = __builtin_nontemporal_load(&rp4[c * 32 + lane]);

  // Row max (lane-local then 5-step xor tree across wave32).
  float m = x[0].x;
#pragma unroll
  for (int c = 0; c < 8; ++c) {
    m = fmaxf(m, fmaxf(fmaxf(x[c].x, x[c].y), fmaxf(x[c].z, x[c].w)));
  }
#pragma unroll
  for (int off = 16; off; off >>= 1)
    m = fmaxf(m, __shfl_xor(m, off, 32));

  // Lane-local sum of exp(x - m).
  float s = 0.0f;
#pragma unroll
  for (int c = 0; c < 8; ++c) {
    s += __expf(x[c].x - m);
    s += __expf(x[c].y - m);
    s += __expf(x[c].z - m);
    s += __expf(x[c].w - m);
  }

  // Cross-lane sum via the matrix unit: D = A x ones(4x16) + 0.
  // A(16x4) layout: lanes 0-15 -> (M=lane, K=0/K=1), lanes 16-31 -> (M=lane-16, K=2/K=3).
  // Put partial in K=0/K=2 slot: D[m,n] = s_m + s_{m+16}.
  // Sum of D's 8 VGPRs gives lanes0-15: sum_{m=0..7} D[m,*], lanes16-31: m=8..15;
  // one xor-16 add completes the 32-lane reduction.
  v2f a;  a.x = s;    a.y = 0.0f;
  v2f bo; bo.x = 1.0f; bo.y = 1.0f;
  v8f cz = {0.f, 0.f, 0.f, 0.f, 0.f, 0.f, 0.f, 0.f};
  v8f d = __builtin_amdgcn_wmma_f32_16x16x4_f32(false, a, false, bo,
                                                (short)0, cz, false, false);
  float tot = d[0] + d[1] + d[2] + d[3] + d[4] + d[5] + d[6] + d[7];
  tot += __shfl_xor(tot, 16, 32);

  const float lse = m + __logf(tot);

  if (lane == 0) {
    blank_lp[row] = x[0].x - lse;          // col 0 lives in lane0 chunk0 .x
    const int u = row % RU1;
    if (u < RU) {
      const int b  = row / (RT * RU1);
      const int yy = y[b * RU + u];
      label_lp[row] = rp[yy] - lse;        // single re-gather; row hot in cache
    }
  }
}

// ---------------------------------------------------------------------------
// Phase 2: alpha recursion. Thread u owns lattice column u; anti-diagonal d=t+u
// steps need only (own prev, left-neighbor prev) -> LDS exchange, 255 steps.
//   alpha[t,u] = logaddexp(alpha[t-1,u]+blank[t-1,u], alpha[t,u-1]+label[t,u-1])
// ---------------------------------------------------------------------------
__global__ __launch_bounds__(RU1)
void rnnt_alpha_kernel(const float* __restrict__ blank_lp,
                       const float* __restrict__ label_lp,
                       const int* __restrict__ logit_lens,
                       const int* __restrict__ y_lens,
                       float* __restrict__ ll) {
  const int b = blockIdx.x;
  const int u = threadIdx.x;  // 0..63
  __shared__ float sh[RU1];

  const float* bl = blank_lp + (size_t)b * RT * RU1;
  const float* la = label_lp + (size_t)b * RT * RU1;
  const int t_tgt = logit_lens[b] - 1;
  const int u_tgt = y_lens[b];

  const float NEG_INF = -__builtin_inff();
  sh[u] = NEG_INF;
  float mine = NEG_INF;
  __syncthreads();

  for (int d = 0; d <= RT - 1 + RU1 - 1; ++d) {
    const int t = d - u;
    const float left = (u > 0) ? sh[u - 1] : NEG_INF;  // = alpha[t][u-1]
    __syncthreads();
    if (t >= 0 && t < RT) {
      float a;
      if (t == 0) {
        a = (u == 0) ? 0.0f : left + la[u - 1];
      } else if (u == 0) {
        a = mine + bl[(t - 1) * RU1];
      } else {
        const float p  = mine + bl[(t - 1) * RU1 + u];
        const float q  = left + la[t * RU1 + (u - 1)];
        const float mx = fmaxf(p, q);
        a = mx + __logf(__expf(p - mx) + __expf(q - mx));
      }
      mine  = a;
      sh[u] = a;
      if (t == t_tgt && u == u_tgt)
        ll[b] = a + bl[t * RU1 + u];       // exactly one thread/step matches
    }
    __syncthreads();
  }
}

// ---------------------------------------------------------------------------
// Phase 3: out = -mean(ll)
// ---------------------------------------------------------------------------
__global__ __launch_bounds__(32)
void rnnt_final_kernel(const float* __restrict__ ll, float* __restrict__ out) {
  const int lane = threadIdx.x;
  float v = (lane < RB) ? ll[lane] : 0.0f;
#pragma unroll
  for (int off = 16; off; off >>= 1)
    v += __shfl_xor(v, off, 32);
  if (lane == 0) out[0] = -v / (float)RB;
}

extern "C" void kernel_launch(void* const* d_in, const int* in_sizes, int n_in,
                              void* d_out, int out_size, void* d_ws, size_t ws_size,
                              hipStream_t stream) {
  const float* logits     = (const float*)d_in[0];
  const int*   logit_lens = (const int*)d_in[1];
  const int*   y          = (const int*)d_in[2];
  const int*   y_lens     = (const int*)d_in[3];
  float*       out        = (float*)d_out;

  float* ws_blank = (float*)d_ws;                       // [B,T,U1]
  float* ws_label = ws_blank + (size_t)RB * RT * RU1;   // [B,T,U1] (u<U valid)
  float* ws_ll    = ws_label + (size_t)RB * RT * RU1;   // [B]

  const int rows = RB * RT * RU1;                       // 196608, 8 waves/block
  rnnt_lse_kernel<<<rows / 8, 256, 0, stream>>>(logits, y, ws_blank, ws_label);
  rnnt_alpha_kernel<<<RB, RU1, 0, stream>>>(ws_blank, ws_label, logit_lens,
                                            y_lens, ws_ll);
  rnnt_final_kernel<<<1, 32, 0, stream>>>(ws_ll, out);
}